// MHSP_54528904790278
// MI455X (gfx1250) — compile-verified
//
#include <hip/hip_runtime.h>
#include <stdint.h>

// Reference: x is (16, 64, 96, 96) f32, NUM_HEAD=4 ->
// out[b, h*136+t, p] = x[b, iu[t]*4+h, p] * x[b, ju[t]*4+h, p]
#define NH      4
#define N16     16
#define TPAIR   136            // 16*17/2 upper-tri pairs
#define OC      (NH * TPAIR)   // 544
#define HW      96
#define LPIX    (HW * HW)      // 9216
#define CIN     64
#define BATCH   16
#define TILE    512            // pixels staged per block (2 per thread)
#define THREADS 256

typedef unsigned int u32x4 __attribute__((ext_vector_type(4)));
typedef unsigned int u32x8 __attribute__((ext_vector_type(8)));

__global__ __launch_bounds__(THREADS)
void mhsp_kernel(const float* __restrict__ x, float* __restrict__ out) {
    __shared__ float smem[CIN * TILE];          // 128 KB: [channel][pixel]

    const int tid = threadIdx.x;
    const int tilesPerBatch = LPIX / TILE;      // 18
    const int b  = blockIdx.x / tilesPerBatch;
    const int l0 = (blockIdx.x % tilesPerBatch) * TILE;

    const float* xb = x + (size_t)b * CIN * LPIX + l0;

    // ---- Path 1 (TDM): one tensor_load_to_lds stages channels [0,32):
    //      2-D tile, 32 rows x 512 f32, row stride LPIX f32. Issued by wave 0
    //      only (TDM ignores EXEC); tracked by TENSORcnt.
    if (tid < 32) {                              // exactly wave 0 (wave32)
        uint64_t ga = (uint64_t)(uintptr_t)xb;
        unsigned la = (unsigned)(uintptr_t)(&smem[0]);
        u32x4 g0;
        g0[0] = 1u;                              // count=1 (valid descriptor)
        g0[1] = la;                              // lds_addr (bytes)
        g0[2] = (unsigned)ga;                    // global_addr[31:0]
        g0[3] = ((unsigned)(ga >> 32) & 0x01FFFFFFu) | (2u << 30); // [56:32] | type=2
        u32x8 g1;
        g1[0] = (2u << 16);                      // workgroup_mask=0, data_size=4B
        g1[1] = ((unsigned)TILE) << 16;          // tensor_dim0[15:0]=512
        g1[2] = (32u << 16);                     // tensor_dim0[31:16]=0 | tensor_dim1[15:0]=32
        g1[3] = ((unsigned)TILE) << 16;          // tensor_dim1[31:16]=0 | tile_dim0=512
        g1[4] = 32u;                             // tile_dim1=32, tile_dim2=0
        g1[5] = (unsigned)LPIX;                  // tensor_dim0_stride[31:0]=9216
        g1[6] = 0u;                              // stride hi / tensor_dim1_stride lo
        g1[7] = 0u;                              // tensor_dim1_stride hi
        asm volatile("tensor_load_to_lds %0, %1"
                     :: "s"(g0), "s"(g1)
                     : "memory");
    }

    // ---- Path 2 (async VMEM): channels [32,64) via per-lane b128 async
    //      copies: 4096 x 16B chunks (128 per channel row), 16 per thread.
    #pragma unroll
    for (int it = 0; it < 16; ++it) {
        int q   = it * THREADS + tid;
        int c   = 32 + (q >> 7);
        int o16 = q & 127;                       // 16B chunk within channel row
        uint64_t gaddr = (uint64_t)(uintptr_t)(xb + (size_t)c * LPIX + o16 * 4);
        unsigned laddr = (unsigned)(uintptr_t)(&smem[c * TILE + o16 * 4]);
        asm volatile("global_load_async_to_lds_b128 %0, %1, off"
                     :: "v"(laddr), "v"(gaddr)
                     : "memory");
    }
    asm volatile("s_wait_asynccnt 0" ::: "memory");
    if (tid < 32) __builtin_amdgcn_s_wait_tensorcnt(0);
    __syncthreads();

    // ---- Compute: per head gather 16 channel pairs from LDS, emit the 136
    //      unique products as coalesced b64 stores (imm offsets <= 4.97 MB).
    for (int head = 0; head < NH; ++head) {
        float2 v[N16];
        #pragma unroll
        for (int j = 0; j < N16; ++j)
            v[j] = *(const float2*)&smem[(j * NH + head) * TILE + 2 * tid];

        float* op = out + ((size_t)b * OC + head * TPAIR) * LPIX + l0 + 2 * tid;
        int t = 0;
        #pragma unroll
        for (int i = 0; i < N16; ++i) {
            #pragma unroll
            for (int j = i; j < N16; ++j) {
                float2 r;
                r.x = v[i].x * v[j].x;
                r.y = v[i].y * v[j].y;
                *(float2*)&op[(size_t)t * LPIX] = r;
                ++t;
            }
        }
    }
}

extern "C" void kernel_launch(void* const* d_in, const int* in_sizes, int n_in,
                              void* d_out, int out_size, void* d_ws, size_t ws_size,
                              hipStream_t stream) {
    const float* x = (const float*)d_in[0];
    float* out = (float*)d_out;

    dim3 grid(BATCH * (LPIX / TILE));   // 16 * 18 = 288 workgroups
    dim3 block(THREADS);                // 8 wave32 waves
    hipLaunchKernelGGL(mhsp_kernel, grid, block, 0, stream, x, out);
}